// TrafficModel_41669772706074
// MI455X (gfx1250) — compile-verified
//
#include <hip/hip_runtime.h>

typedef __attribute__((ext_vector_type(2))) float v2f;
typedef __attribute__((ext_vector_type(8))) float v8f;

#define NL 3000      // links
#define NP 17700     // paths
#define NO 3540      // OD pairs
#define KP 5         // paths per OD
#define NCH_U 6      // link chunks for u-pass
#define CH_U 500     // NCH_U * CH_U == NL
#define NCH_X 4      // path chunks for x-pass
#define CH_X 4428    // multiple of 4; NCH_X * CH_X >= NP

// ---------------- kernel 1: BPR link travel times ----------------
__global__ void bpr_time_kernel(const float* __restrict__ x_hat,
                                const float* __restrict__ alpha,
                                const float* __restrict__ beta,
                                const float* __restrict__ t_min,
                                const float* __restrict__ x_max,
                                float* __restrict__ t_out,
                                float* __restrict__ t_ws) {
  int l = blockIdx.x * blockDim.x + threadIdx.x;
  if (l < NL) {
    float base = 1.0f + alpha[l] * (x_hat[l] / x_max[l]);   // base >= ~1
    float t = t_min[l] * __powf(base, beta[l]);
    t_out[l] = t;
    t_ws[l] = t;
  }
}

// ---------------- kernel 2: u-partials = D^T t via WMMA f32 16x16x4 ----------
// Each wave owns 16 consecutive paths. A[16x4] = D^T tile:
//   lane m = lane&15 selects path p0+m (consecutive -> coalesced 64B rows),
//   K striping: VGPR0 holds K = (lane<16 ? 0 : 2), VGPR1 holds K+1.
// B[4x16] = t broadcast to all 16 columns, same K striping -> every result
// column equals the u-partial for that path; extract from lanes 0 and 16.
__global__ void u_partial_kernel(const float* __restrict__ D,
                                 const float* __restrict__ t,
                                 float* __restrict__ upart) {
  __shared__ float ts[CH_U];
  const int lane = threadIdx.x & 31;
  const int wave = threadIdx.x >> 5;
  const int chunk = blockIdx.y;
  const int l0 = chunk * CH_U;

  for (int i = threadIdx.x; i < CH_U; i += blockDim.x) ts[i] = t[l0 + i];
  __syncthreads();

  const int p0 = blockIdx.x * 128 + wave * 16;
  const int m  = lane & 15;
  const int kh = (lane >> 4) << 1;                 // 0 or 2
  int pm = p0 + m;
  if (pm >= NP) pm = NP - 1;                       // clamp tail tile (uniform loop, EXEC all 1s)

  const float* __restrict__ pa = D + (size_t)(l0 + kh) * NP + pm;
  const float* __restrict__ pb = ts + kh;

  v8f c = {};
  for (int it = 0; it < CH_U / 4; ++it) {
    v2f a, b;
    a.x = pa[0];                                   // K = kh
    a.y = pa[NP];                                  // K = kh+1
    b.x = pb[0];                                   // t[l0+kh]   (b64 from LDS)
    b.y = pb[1];                                   // t[l0+kh+1]
    c = __builtin_amdgcn_wmma_f32_16x16x4_f32(false, a, false, b,
                                              (short)0, c, false, false);
    pa += (size_t)4 * NP;
    pb += 4;
  }

  // C/D layout: VGPR v -> M=v (lanes 0-15) / M=v+8 (lanes 16-31); all N equal.
  if (m == 0) {
    const int base = (lane >> 4) * 8;
#pragma unroll
    for (int v = 0; v < 8; ++v) {
      int pp = p0 + base + v;
      if (pp < NP) upart[(size_t)chunk * NP + pp] = c[v];
    }
  }
}

// ---------------- kernel 3: reduce u-partials + segmented softmax + demand ---
__global__ void softmax_kernel(const float* __restrict__ upart,
                               const float* __restrict__ q_hat,
                               float* __restrict__ p_out,
                               float* __restrict__ f_out,
                               float* __restrict__ f_ws) {
  int o = blockIdx.x * blockDim.x + threadIdx.x;
  if (o >= NO) return;
  float u[KP];
#pragma unroll
  for (int j = 0; j < KP; ++j) {
    float s = 0.0f;
#pragma unroll
    for (int c = 0; c < NCH_U; ++c) s += upart[(size_t)c * NP + o * KP + j];
    u[j] = s;
  }
  float mx = u[0];
#pragma unroll
  for (int j = 1; j < KP; ++j) mx = fmaxf(mx, u[j]);
  float e[KP], s = 0.0f;
#pragma unroll
  for (int j = 0; j < KP; ++j) { e[j] = __expf(u[j] - mx); s += e[j]; }
  float inv = 1.0f / s;
  float q = q_hat[o];
#pragma unroll
  for (int j = 0; j < KP; ++j) {
    float pj = e[j] * inv;
    float fj = q * pj;
    p_out[o * KP + j] = pj;
    f_out[o * KP + j] = fj;
    f_ws[o * KP + j] = fj;
  }
}

// ---------------- kernel 4: x-partials = D f (row streaming + wave reduce) ---
__global__ void x_partial_kernel(const float* __restrict__ D,
                                 const float* __restrict__ f,
                                 float* __restrict__ xpart) {
  const int lane = threadIdx.x & 31;
  const int wave = threadIdx.x >> 5;
  const int l = blockIdx.x * 8 + wave;             // 375 * 8 == NL exactly
  const int chunk = blockIdx.y;
  const int pstart = chunk * CH_X;
  const int pend = min(NP, pstart + CH_X);         // both multiples of 4

  const float* __restrict__ row = D + (size_t)l * NP;
  float acc = 0.0f;
  for (int p = pstart + lane * 4; p < pend; p += 32 * 4) {
    float4 d4 = *reinterpret_cast<const float4*>(row + p);  // NP%4==0 -> 16B aligned
    float4 f4 = *reinterpret_cast<const float4*>(f + p);
    acc = fmaf(d4.x, f4.x, acc);
    acc = fmaf(d4.y, f4.y, acc);
    acc = fmaf(d4.z, f4.z, acc);
    acc = fmaf(d4.w, f4.w, acc);
  }
#pragma unroll
  for (int off = 16; off > 0; off >>= 1)
    acc += __shfl_xor(acc, off, 32);
  if (lane == 0) xpart[(size_t)chunk * NL + l] = acc;
}

// ---------------- kernel 5: deterministic final sum for x --------------------
__global__ void x_final_kernel(const float* __restrict__ xpart,
                               float* __restrict__ x_out) {
  int l = blockIdx.x * blockDim.x + threadIdx.x;
  if (l < NL) {
    float s = 0.0f;
#pragma unroll
    for (int c = 0; c < NCH_X; ++c) s += xpart[(size_t)c * NL + l];
    x_out[l] = s;
  }
}

extern "C" void kernel_launch(void* const* d_in, const int* in_sizes, int n_in,
                              void* d_out, int out_size, void* d_ws, size_t ws_size,
                              hipStream_t stream) {
  // setup_inputs order: x_hat, alpha, beta, q_hat, D, M, t_min, x_max
  const float* x_hat = (const float*)d_in[0];
  const float* alpha = (const float*)d_in[1];
  const float* beta  = (const float*)d_in[2];
  const float* q_hat = (const float*)d_in[3];
  const float* D     = (const float*)d_in[4];
  // d_in[5] (M) is one-hot with seg = p // K; never materialized.
  const float* t_min = (const float*)d_in[6];
  const float* x_max = (const float*)d_in[7];

  float* out   = (float*)d_out;
  float* out_x = out;                  // [NL]
  float* out_t = out + NL;             // [NL]
  float* out_f = out + 2 * NL;         // [NP]
  float* out_p = out + 2 * NL + NP;    // [NP]

  float* ws    = (float*)d_ws;
  float* ws_t  = ws;                               // [NL]
  float* ws_up = ws_t + NL;                        // [NCH_U * NP]
  float* ws_f  = ws_up + (size_t)NCH_U * NP;       // [NP]  (offset 109200, 16B aligned)
  float* ws_xp = ws_f + NP;                        // [NCH_X * NL]

  bpr_time_kernel<<<(NL + 255) / 256, 256, 0, stream>>>(
      x_hat, alpha, beta, t_min, x_max, out_t, ws_t);

  dim3 gu((NP + 127) / 128, NCH_U);                // 139 x 6 blocks, 8 waves each
  u_partial_kernel<<<gu, 256, 0, stream>>>(D, ws_t, ws_up);

  softmax_kernel<<<(NO + 255) / 256, 256, 0, stream>>>(
      ws_up, q_hat, out_p, out_f, ws_f);

  dim3 gx(NL / 8, NCH_X);                          // 375 x 4 blocks
  x_partial_kernel<<<gx, 256, 0, stream>>>(D, ws_f, ws_xp);

  x_final_kernel<<<(NL + 255) / 256, 256, 0, stream>>>(ws_xp, out_x);
}